// PhotogrammetricRenderer_47536698032161
// MI455X (gfx1250) — compile-verified
//
#include <hip/hip_runtime.h>
#include <stdint.h>

// ---------------------------------------------------------------------------
// NeRF volume rendering, MI455X (gfx1250, wave32).
// One wave per ray; 128 samples = 4 chunks of 32 lanes.
// Exclusive cumprod via wave32 prefix-product scan, carry chained per chunk.
// Inputs staged through LDS with gfx1250 async copies (ASYNCcnt), double
// buffered so chunk c+1 streams in while chunk c is computed.
// ---------------------------------------------------------------------------

#define NSAMP          128
#define CHUNK          32
#define NCHUNK         (NSAMP / CHUNK)
#define WAVES_PER_BLK  8
#define BLOCK_THREADS  (WAVES_PER_BLK * 32)

// Per-buffer dword layout: [0,32) densities | [32,64) z | [64,160) colors AoS
#define BUF_DW         160

// gfx1250 async global->LDS copy, GV mode (64-bit VGPR address, per-lane).
// VDST VGPR holds the workgroup-relative LDS byte address (low 32 bits of the
// generic pointer to __shared__, per ISA flat->LDS truncation rule).
__device__ __forceinline__ void async_ld_b32(uint32_t lds_byte_off, const void* gaddr) {
    asm volatile("global_load_async_to_lds_b32 %0, %1, off"
                 :: "v"(lds_byte_off), "v"(gaddr)
                 : "memory");
}

template <int N>
__device__ __forceinline__ void wait_asynccnt() {
#if __has_builtin(__builtin_amdgcn_s_wait_asynccnt)
    __builtin_amdgcn_s_wait_asynccnt(N);
#else
    asm volatile("s_wait_asynccnt %0" :: "i"(N) : "memory");
#endif
}

__global__ __launch_bounds__(BLOCK_THREADS)
void PhotogrammetricRenderer_47536698032161_kernel(
    const float* __restrict__ densities,   // [N, S, 1]
    const float* __restrict__ colors,      // [N, S, 3]
    const float* __restrict__ z_vals,      // [N, S]
    const float* __restrict__ rays_d,      // [N, 3]
    const int*   __restrict__ white_bkgd,  // scalar
    float*       __restrict__ out,         // rgb|depth|disp|acc|weights
    int n_rays)
{
    __shared__ float smem[WAVES_PER_BLK][2][BUF_DW];

    const int lane = threadIdx.x & 31;
    const int wave = threadIdx.x >> 5;
    const int ray  = blockIdx.x * WAVES_PER_BLK + wave;
    if (ray >= n_rays) return;                    // wave-uniform

    const long long base = (long long)ray * NSAMP;
    float* sm = &smem[wave][0][0];                // two buffers: sm, sm+BUF_DW

    // Issue the 5 async copies for one chunk into one LDS buffer.
    auto issue = [&](int c, int bufsel) {
        const int s = c * CHUNK + lane;
        const uint32_t lbase =
            (uint32_t)(uintptr_t)(sm + bufsel * BUF_DW);
        async_ld_b32(lbase + (uint32_t)lane * 4u,           densities + base + s);
        async_ld_b32(lbase + 128u + (uint32_t)lane * 4u,    z_vals    + base + s);
        const float* cp = colors + (base + s) * 3;
        const uint32_t cl = lbase + 256u + (uint32_t)lane * 12u;
        async_ld_b32(cl + 0u, cp + 0);
        async_ld_b32(cl + 4u, cp + 1);
        async_ld_b32(cl + 8u, cp + 2);
    };

    // Ray direction norm (broadcast loads: all lanes same address).
    const float rd0 = rays_d[ray * 3 + 0];
    const float rd1 = rays_d[ray * 3 + 1];
    const float rd2 = rays_d[ray * 3 + 2];
    const float norm = sqrtf(rd0 * rd0 + rd1 * rd1 + rd2 * rd2);

    issue(0, 0);

    float carry = 1.0f;                 // running exclusive-cumprod carry
    float aW = 0.f, aD = 0.f, aR = 0.f, aG = 0.f, aB = 0.f;

    #pragma unroll
    for (int c = 0; c < NCHUNK; ++c) {
        const int bufsel = c & 1;
        if (c + 1 < NCHUNK) {           // prefetch next chunk, wait on current
            issue(c + 1, bufsel ^ 1);
            wait_asynccnt<5>();         // next chunk's 5 ops still in flight
        } else {
            wait_asynccnt<0>();
        }

        const float* buf = sm + bufsel * BUF_DW;
        const int s = c * CHUNK + lane;

        const float dens = buf[lane];
        const float z    = buf[32 + lane];
        const float c0   = buf[64 + lane * 3 + 0];
        const float c1   = buf[64 + lane * 3 + 1];
        const float c2   = buf[64 + lane * 3 + 2];

        // dists: z[s+1]-z[s]; last interval = 1e10; scaled by |rays_d|
        float zn = __shfl_down(z, 1, 32);
        if (lane == 31 && s < NSAMP - 1) zn = z_vals[base + s + 1]; // L2 hit
        float dist = (s == NSAMP - 1) ? 1e10f : (zn - z);
        dist *= norm;

        const float sigma = fmaxf(dens, 0.0f);
        const float alpha = 1.0f - __expf(-sigma * dist);
        const float t     = 1.0f - alpha + 1e-10f;

        // wave32 inclusive prefix product of t
        float p = t;
        #pragma unroll
        for (int off = 1; off < 32; off <<= 1) {
            const float y = __shfl_up(p, off, 32);
            if (lane >= off) p *= y;
        }
        // exclusive within chunk
        float excl = __shfl_up(p, 1, 32);
        if (lane == 0) excl = 1.0f;

        const float w = alpha * (carry * excl);
        carry *= __shfl(p, 31, 32);     // whole-chunk product

        // weights output: pure streaming, bypass cache
        __builtin_nontemporal_store(w, out + 6ll * n_rays + base + s);

        aW += w;
        aD += w * z;
        aR += w * c0;
        aG += w * c1;
        aB += w * c2;
    }

    // Butterfly reduction across the 32 lanes.
    #pragma unroll
    for (int off = 16; off >= 1; off >>= 1) {
        aW += __shfl_xor(aW, off, 32);
        aD += __shfl_xor(aD, off, 32);
        aR += __shfl_xor(aR, off, 32);
        aG += __shfl_xor(aG, off, 32);
        aB += __shfl_xor(aB, off, 32);
    }

    if (lane == 0) {
        const int wb = *white_bkgd;
        if (wb) {
            const float bg = 1.0f - aW;
            aR += bg; aG += bg; aB += bg;
        }
        out[(long long)ray * 3 + 0] = aR;
        out[(long long)ray * 3 + 1] = aG;
        out[(long long)ray * 3 + 2] = aB;
        out[3ll * n_rays + ray] = aD;                              // depth
        out[4ll * n_rays + ray] = 1.0f / fmaxf(1e-10f, aD / aW);   // disp
        out[5ll * n_rays + ray] = aW;                              // acc_alpha
    }
}

extern "C" void kernel_launch(void* const* d_in, const int* in_sizes, int n_in,
                              void* d_out, int out_size, void* d_ws, size_t ws_size,
                              hipStream_t stream) {
    const float* densities = (const float*)d_in[0];
    const float* colors    = (const float*)d_in[1];
    const float* z_vals    = (const float*)d_in[2];
    const float* rays_d    = (const float*)d_in[3];
    const int*   white     = (const int*)d_in[4];
    float* out = (float*)d_out;

    const int n_rays = in_sizes[2] / NSAMP;   // z_vals is [N, S]
    const int blocks = (n_rays + WAVES_PER_BLK - 1) / WAVES_PER_BLK;

    PhotogrammetricRenderer_47536698032161_kernel<<<blocks, BLOCK_THREADS, 0, stream>>>(
        densities, colors, z_vals, rays_d, white, out, n_rays);
}